// SemanticDependencyMFVI_5652176961692
// MI455X (gfx1250) — compile-verified
//
#include <hip/hip_runtime.h>
#include <hip/hip_bf16.h>

typedef float v2f __attribute__((ext_vector_type(2)));
typedef float v8f __attribute__((ext_vector_type(8)));

#define BSZ 8
#define LSZ 128
#define NEL (BSZ * LSZ * LSZ)   // 131072 elements of [B,L,L]
#define ROWP 130                // padded LDS row stride (floats) -> conflict-free

// prep: float mask copy + p = sigmoid(s_edge)   (q starts as s_edge)
__global__ void prep_kernel(const float* __restrict__ s_edge,
                            const unsigned char* __restrict__ mask,
                            float* __restrict__ p, float* __restrict__ Mf) {
  int i = blockIdx.x * blockDim.x + threadIdx.x;
  if (i < NEL) {
    Mf[i] = mask[i] ? 1.0f : 0.0f;
    p[i] = 1.0f / (1.0f + __expf(-s_edge[i]));
  }
}

__global__ void sigmoid_kernel(const float* __restrict__ src,
                               float* __restrict__ dst) {
  int i = blockIdx.x * blockDim.x + threadIdx.x;
  if (i < NEL) {
    dst[i] = 1.0f / (1.0f + __expf(-src[i]));
  }
}

// One block = 8 waves = 8 consecutive h values of one (b, 16-wide m-tile).
// Shared LDS tiles give 8x reuse of p/Mf; each wave streams its three tensor
// rows with b64 loads and reduces over s via 96 chained V_WMMA_F32_16X16X4_F32.
// The (s != h) exclusion is applied as a scalar correction after the loop.
__global__ __launch_bounds__(256)
void mfvi_kernel(const float* __restrict__ s_edge,
                 const float* __restrict__ s_sib,
                 const float* __restrict__ s_cop,
                 const float* __restrict__ s_grd,
                 const float* __restrict__ p,
                 const float* __restrict__ Mf,
                 float* __restrict__ q_out) {
  __shared__ float PM1[16 * ROWP];   // p[s,m]*Mf[s,m]*(s!=m), layout [m][s]
  __shared__ float MF2[16 * ROWP];   // Mf[s,m]*(s!=m),        layout [m][s]
  __shared__ float PROW[8][LSZ];     // p[b,h,s]  per wave
  __shared__ float PCOL[8][LSZ];     // p[b,s,h]  per wave

  const int tid  = threadIdx.x;
  const int lane = tid & 31;
  const int wv   = tid >> 5;                 // wave in block = h offset
  const int hg   = blockIdx.x & 15;          // h-group (8 h each)
  const int mt   = (blockIdx.x >> 4) & 7;    // m-tile
  const int b    = blockIdx.x >> 7;          // batch
  const int m0   = mt * 16;
  const int h    = hg * 8 + wv;

  const float* pb = p  + (size_t)b * LSZ * LSZ;
  const float* mb = Mf + (size_t)b * LSZ * LSZ;

  // ---- stage PM1 / MF2 tiles (once per block, coalesced 64B per 16 threads)
  {
    const int ml = tid & 15;
    const int sb = tid >> 4;                 // 0..15
    const int mg = m0 + ml;
#pragma unroll
    for (int rep = 0; rep < 8; ++rep) {
      const int s = sb + rep * 16;
      float mf = mb[s * LSZ + mg];
      float pv = pb[s * LSZ + mg];
      mf = (s != mg) ? mf : 0.0f;
      PM1[ml * ROWP + s] = pv * mf;
      MF2[ml * ROWP + s] = mf;
    }
  }
  // ---- stage per-wave p row / column
  {
    const float4 pr = *(const float4*)(pb + h * LSZ + lane * 4);
    *(float4*)(&PROW[wv][lane * 4]) = pr;
    float4 pc;
    pc.x = pb[(lane * 4 + 0) * LSZ + h];
    pc.y = pb[(lane * 4 + 1) * LSZ + h];
    pc.z = pb[(lane * 4 + 2) * LSZ + h];
    pc.w = pb[(lane * 4 + 3) * LSZ + h];
    *(float4*)(&PCOL[wv][lane * 4]) = pc;
  }
  __syncthreads();

  const int n  = lane & 15;   // A-row M and B-col N within tile
  const int kg = lane >> 4;   // K-group (K = 2*kg + v)
  const int m  = m0 + n;

  const size_t bhm = (size_t)(b * LSZ + h) * LSZ + m;
  const float* sibRow = s_sib + bhm * LSZ + 2 * kg;
  const float* copRow = s_cop + bhm * LSZ + 2 * kg;
  const float* grdRow = s_grd + bhm * LSZ + 2 * kg;
  const float* pm1p = &PM1[n * ROWP + 2 * kg];
  const float* mf2p = &MF2[n * ROWP + 2 * kg];
  const float* prp  = &PROW[wv][2 * kg];
  const float* pcp  = &PCOL[wv][2 * kg];

  v8f acc = {};
#pragma unroll
  for (int ch = 0; ch < LSZ / 4; ++ch) {
    const int o = ch * 4;                       // constant after full unroll
    v2f a1  = *(const v2f*)(sibRow + o);        // HBM streams (b64, coalesced)
    v2f a2  = *(const v2f*)(copRow + o);
    v2f a3  = *(const v2f*)(grdRow + o);
    v2f b1  = *(const v2f*)(pm1p + o);          // ds_load_b64, imm offsets
    v2f mf2 = *(const v2f*)(mf2p + o);
    v2f pr  = *(const v2f*)(prp + o);
    v2f pc  = *(const v2f*)(pcp + o);
    v2f b2 = pr * mf2;
    v2f b3 = pc * mf2;
    acc = __builtin_amdgcn_wmma_f32_16x16x4_f32(false, a1, false, b1,
                                                (short)0, acc, false, false);
    acc = __builtin_amdgcn_wmma_f32_16x16x4_f32(false, a2, false, b2,
                                                (short)0, acc, false, false);
    acc = __builtin_amdgcn_wmma_f32_16x16x4_f32(false, a3, false, b3,
                                                (short)0, acc, false, false);
  }

  // ---- diagonal extraction: C(M=mloc,N=mloc) sits in VGPR (mloc&7),
  //      lanes 0..7 (M=0..7) and lanes 24..31 (M=8..15).
  const bool active = (lane < 8) || (lane >= 24);
  if (active) {
    const int mloc = (lane < 8) ? lane : (lane - 16);
    const int r = mloc & 7;
    float dsum = acc[0];
#pragma unroll
    for (int rr = 1; rr < 8; ++rr)
      if (r == rr) dsum = acc[rr];

    const int mo = m0 + mloc;
    const size_t oidx = (size_t)(b * LSZ + h) * LSZ + mo;

    // subtract the s == h summand (the (s != h) mask term)
    const size_t rbase = oidx * LSZ + h;
    const float sib_h = s_sib[rbase];
    const float cop_h = s_cop[rbase];
    const float grd_h = s_grd[rbase];
    const float pm1_h = PM1[mloc * ROWP + h];
    const float mf2_h = MF2[mloc * ROWP + h];   // == Mf[b,h,mo]*(h!=mo) == cf
    const float p_hh  = PROW[wv][h];            // p[b,h,h]
    const float corr  = pm1_h * sib_h + mf2_h * p_hh * (cop_h + grd_h);

    q_out[oidx] = s_edge[oidx] + mf2_h * (dsum - corr);
  }
}

extern "C" void kernel_launch(void* const* d_in, const int* in_sizes, int n_in,
                              void* d_out, int out_size, void* d_ws, size_t ws_size,
                              hipStream_t stream) {
  (void)in_sizes; (void)n_in; (void)out_size; (void)ws_size;
  const float* s_edge = (const float*)d_in[0];
  const float* s_sib  = (const float*)d_in[1];
  const float* s_cop  = (const float*)d_in[2];
  const float* s_grd  = (const float*)d_in[3];
  const unsigned char* mask = (const unsigned char*)d_in[4];
  float* out = (float*)d_out;

  float* p  = (float*)d_ws;      // [NEL]
  float* q  = p + NEL;           // [NEL]
  float* Mf = q + NEL;           // [NEL]

  const dim3 eb(256), eg((NEL + 255) / 256);
  const dim3 mblk(256), mgrd(BSZ * 8 * 16);   // 1024 blocks: (b, m-tile, h-group)

  prep_kernel<<<eg, eb, 0, stream>>>(s_edge, mask, p, Mf);

  // iteration 1
  mfvi_kernel<<<mgrd, mblk, 0, stream>>>(s_edge, s_sib, s_cop, s_grd, p, Mf, q);
  sigmoid_kernel<<<eg, eb, 0, stream>>>(q, p);
  // iteration 2
  mfvi_kernel<<<mgrd, mblk, 0, stream>>>(s_edge, s_sib, s_cop, s_grd, p, Mf, q);
  sigmoid_kernel<<<eg, eb, 0, stream>>>(q, p);
  // iteration 3
  mfvi_kernel<<<mgrd, mblk, 0, stream>>>(s_edge, s_sib, s_cop, s_grd, p, Mf, q);
  sigmoid_kernel<<<eg, eb, 0, stream>>>(q, out);
}